// QuantizedEmbedding_34016140984681
// MI455X (gfx1250) — compile-verified
//
#include <hip/hip_runtime.h>
#include <stdint.h>

// QuantizedEmbedding gather-dequant for MI455X (gfx1250).
//
// out[t, :] = (float(qweight[idx[t], :]) - zeros[idx[t]]) * scales[idx[t]]
//
// Pure HBM-bound row gather (128 MB total traffic -> ~5.5us at 23.3 TB/s).
// Uses the CDNA5 Tensor Data Mover in GATHER MODE: one TENSOR_LOAD_TO_LDS
// per block DMAs 8 arbitrary 4KB rows (32-bit row indices in D# groups 2/3)
// directly into LDS, bypassing VGPRs. Waves then dequantize from LDS with
// ds_load_b128 and stream results out with nontemporal b128 stores.

typedef __attribute__((ext_vector_type(4))) unsigned int v4u;
typedef __attribute__((ext_vector_type(4))) int          v4i;
typedef __attribute__((ext_vector_type(8))) int          v8i;
typedef __attribute__((ext_vector_type(4))) float        v4f;

#define EMB_DIM        1024     // row length (elements)
#define QE_VOCAB       128000   // number of rows
#define ROWS_PER_BLOCK 8        // 8 x 32-bit gather indices per TDM descriptor

__global__ __launch_bounds__(256) void qembed_tdm_gather(
    const int*   __restrict__ input,    // [ntok] token ids
    const int*   __restrict__ qweight,  // [VOCAB, DIM] uint8 codes in int32
    const float* __restrict__ scales,   // [VOCAB]
    const float* __restrict__ zeros,    // [VOCAB]
    float*       __restrict__ out,      // [ntok, DIM]
    int ntok)
{
    // 8 rows x 1024 int32 = 32 KB staged by the TDM.
    __shared__ int rows[ROWS_PER_BLOCK * EMB_DIM];

    const int t0   = (int)blockIdx.x * ROWS_PER_BLOCK;
    const int tid  = (int)threadIdx.x;
    const int wave = tid >> 5;
    const int lane = tid & 31;

    if (wave == 0) {
        // Lanes 0..7 fetch the 8 token ids (lanes 8..31 fetch duplicates so
        // the load stays wave-uniform in control flow).
        int tok = t0 + (lane & 7);
        if (tok >= ntok) tok = ntok - 1;         // tail clamp (harmless dup gather)
        const int myIdx = input[tok];

        const int i0 = __builtin_amdgcn_readlane(myIdx, 0);
        const int i1 = __builtin_amdgcn_readlane(myIdx, 1);
        const int i2 = __builtin_amdgcn_readlane(myIdx, 2);
        const int i3 = __builtin_amdgcn_readlane(myIdx, 3);
        const int i4 = __builtin_amdgcn_readlane(myIdx, 4);
        const int i5 = __builtin_amdgcn_readlane(myIdx, 5);
        const int i6 = __builtin_amdgcn_readlane(myIdx, 6);
        const int i7 = __builtin_amdgcn_readlane(myIdx, 7);

        // LDS byte offset: low 32 bits of the flat address of a __shared__
        // object are the LDS offset (shared aperture base has zero low word).
        const unsigned lds_addr = (unsigned)(uintptr_t)(void*)rows;
        const unsigned long long gaddr = (unsigned long long)(uintptr_t)qweight;

        // ---- D# group 0 (128b): control | lds_addr | global_addr | type ----
        v4u g0;
        g0.x = 1u                  // count = 1 (one valid descriptor)
             | (1u << 30)          // gather_index_size = 1 (32-bit indices)
             | (1u << 31);         // gather_mode = 1
        g0.y = lds_addr;                                   // bits [63:32]
        g0.z = (unsigned)gaddr;                            // global_addr lo
        g0.w = (unsigned)((gaddr >> 32) & 0x01FFFFFFull)   // global_addr[56:32]
             | (2u << 30);                                 // type = 2 ("image")

        // ---- D# group 1 (256b): geometry --------------------------------
        v8i g1;
        g1[0] = (int)(2u << 16);                       // wg_mask=0, data_size=2 (4B)
        g1[1] = (int)((unsigned)(EMB_DIM & 0xFFFF) << 16);            // tensor_dim0[15:0]
        g1[2] = (int)(((unsigned)(QE_VOCAB & 0xFFFF) << 16)           // tensor_dim1[15:0]
                     | ((unsigned)EMB_DIM >> 16));                    // tensor_dim0[31:16]
        g1[3] = (int)(((unsigned)(EMB_DIM & 0xFFFF) << 16)            // tile_dim0 = 1024
                     | ((unsigned)QE_VOCAB >> 16));                   // tensor_dim1[31:16]
        g1[4] = ROWS_PER_BLOCK;                        // tile_dim1 = #valid indices; tile_dim2 = 0
        g1[5] = EMB_DIM;                               // tensor_dim0_stride lo32 (elements)
        g1[6] = 0;                                     // stride hi16 | dim1_stride (ignored)
        g1[7] = 0;

        // ---- D# groups 2/3: 32-bit row indices 0..7 ----------------------
        v4i g2 = { i0, i1, i2, i3 };
        v4i g3 = { i4, i5, i6, i7 };

#if defined(__clang_major__) && (__clang_major__ >= 23)
        v8i gpad = { 0, 0, 0, 0, 0, 0, 0, 0 };
        __builtin_amdgcn_tensor_load_to_lds(g0, g1, g2, g3, gpad, 0);
#else
        __builtin_amdgcn_tensor_load_to_lds(g0, g1, g2, g3, 0);
#endif
        __builtin_amdgcn_s_wait_tensorcnt(0);
    }
    __syncthreads();   // rows[] now holds 8 gathered 4KB rows

    // ---- Dequantize: wave w handles row w -------------------------------
    const int tok = t0 + wave;
    if (tok < ntok) {
        const int   idx = input[tok];     // wave-uniform -> scalar load
        const float s   = scales[idx];
        const float z   = zeros[idx];

        const v4i* __restrict__ src =
            reinterpret_cast<const v4i*>(&rows[wave * EMB_DIM]);
        v4f* __restrict__ dst =
            reinterpret_cast<v4f*>(out + (size_t)tok * EMB_DIM);

        // 256 int4 per row; 32 lanes x 8 chunks, fully coalesced b128s.
        #pragma unroll
        for (int c = 0; c < 8; ++c) {
            const int e4 = c * 32 + lane;
            const v4i q  = src[e4];                  // ds_load_b128
            v4f f;
            f.x = ((float)q.x - z) * s;              // match reference rounding
            f.y = ((float)q.y - z) * s;
            f.z = ((float)q.z - z) * s;
            f.w = ((float)q.w - z) * s;
            __builtin_nontemporal_store(f, &dst[e4]); // global_store_b128 NT
        }
    }
}

extern "C" void kernel_launch(void* const* d_in, const int* in_sizes, int n_in,
                              void* d_out, int out_size, void* d_ws, size_t ws_size,
                              hipStream_t stream) {
    (void)n_in; (void)out_size; (void)d_ws; (void)ws_size;

    const int*   input   = (const int*)  d_in[0];   // [B*S] token ids
    const int*   qweight = (const int*)  d_in[1];   // [VOCAB*DIM]
    const float* scales  = (const float*)d_in[2];   // [VOCAB]
    const float* zeros   = (const float*)d_in[3];   // [VOCAB]
    float*       out     = (float*)      d_out;     // [B*S*DIM]

    const int ntok = in_sizes[0];                   // 8 * 2048 = 16384
    const int nblk = (ntok + ROWS_PER_BLOCK - 1) / ROWS_PER_BLOCK;

    qembed_tdm_gather<<<dim3(nblk), dim3(256), 0, stream>>>(
        input, qweight, scales, zeros, out, ntok);
}